// Quantizer_12902081757269
// MI455X (gfx1250) — compile-verified
//
#include <hip/hip_runtime.h>

typedef __attribute__((ext_vector_type(2))) float v2f;
typedef __attribute__((ext_vector_type(8))) float v8f;
typedef __attribute__((ext_vector_type(4))) int   v4i;

#define B_   8
#define C_   512
#define T_   4096
#define G_   2
#define NC_  1024
#define E_   256
#define N_   (B_ * T_)                 // 32768 vectors per group
#define M_TOT ((size_t)B_ * C_ * T_)   // 16777216 elements
#define ROWS 64                        // t-rows per block tile
#define LDA  260                       // padded LDS stride (260 % 64 == 4)
#define LDB  260
#define CODES_PER_ITER 32              // two 16-wide WMMA sub-tiles
#define NITER (NC_ / CODES_PER_ITER)   // 32

// ---- optional CDNA5 async global->LDS path (compile-safe fallback) --------
#if defined(__has_builtin)
#  if __has_builtin(__builtin_amdgcn_global_load_async_to_lds_b128) && \
      __has_builtin(__builtin_amdgcn_s_wait_asynccnt)
#    define HAVE_ASYNC 1
#  endif
#endif
#ifndef HAVE_ASYNC
#  define HAVE_ASYNC 0
#endif

#if HAVE_ASYNC
#  define ASYNC_COPY16(gp, lp) \
     __builtin_amdgcn_global_load_async_to_lds_b128((v4i*)(gp), (v4i*)(lp), 0, 0)
#  define WAIT_ASYNC(n) __builtin_amdgcn_s_wait_asynccnt(n)
#else
#  define ASYNC_COPY16(gp, lp) (*(float4*)(lp) = *(const float4*)(gp))
#  define WAIT_ASYNC(n) ((void)0)
#endif

// ---------------------------------------------------------------------------
// ||c||^2 per code, one wave per code, both codebooks in one launch.
__global__ void cnorm_kernel(const float* __restrict__ cb1,
                             const float* __restrict__ cb2,
                             float* __restrict__ cn1,
                             float* __restrict__ cn2) {
  int wave = (blockIdx.x * blockDim.x + threadIdx.x) >> 5;
  int lane = threadIdx.x & 31;
  const float* cb = (wave < G_ * NC_) ? cb1 : cb2;
  float*       cn = (wave < G_ * NC_) ? cn1 : cn2;
  int code = wave & (G_ * NC_ - 1);
  const float* row = cb + (size_t)code * E_;
  float s = 0.0f;
  #pragma unroll
  for (int e = lane; e < E_; e += 32) { float v = row[e]; s += v * v; }
  #pragma unroll
  for (int off = 16; off > 0; off >>= 1) s += __shfl_xor(s, off, 32);
  if (lane == 0) cn[code] = s;
}

__global__ void zero_kernel(float* sums) {
  if (threadIdx.x < 2) sums[threadIdx.x] = 0.0f;
}

// ---------------------------------------------------------------------------
// Distance-argmin: for a [ROWS x E] tile of vectors (one group/batch, 64
// consecutive t), find argmin_c ||c||^2 - 2 x.c over NC codes.
// GEMM on V_WMMA_F32_16X16X4_F32, dual accumulator chains per wave, B tile
// double-buffered with async global->LDS prefetch one tile ahead.
// Stage 2 subtracts the stage-1 code during the A-tile load.
__global__ void argmin_kernel(const float* __restrict__ x,
                              const float* __restrict__ cb,        // [G,NC,E]
                              const float* __restrict__ cnorm,     // [G*NC]
                              const float* __restrict__ prev_cb,   // null: stage 1
                              const int*   __restrict__ prev_idx,  // null: stage 1
                              int* __restrict__ idx_out)           // [G*N]
{
  extern __shared__ float smem[];
  float* As = smem;                              // ROWS * LDA
  float* Bs = As + ROWS * LDA;                   // 2 * 32 * LDB (double buffer)
  float* Cn = Bs + 2 * CODES_PER_ITER * LDB;     // NC_
  int*   Pi = (int*)(Cn + NC_);                  // ROWS

  const int bid = blockIdx.x;                    // grid = G_*B_*(T_/ROWS) = 1024
  const int g   = bid >> 9;
  const int rem = bid & 511;
  const int b   = rem >> 6;
  const int tc  = rem & 63;
  const int t0  = tc * ROWS;
  const int tid  = threadIdx.x;                  // 128 threads = 4 waves
  const int lane = tid & 31;
  const int w    = tid >> 5;

  const size_t xbase  = ((size_t)b * C_ + (size_t)g * E_) * T_ + t0;
  const size_t nbase  = (size_t)g * N_ + (size_t)b * T_ + t0;
  const float* cbg    = cb + (size_t)g * NC_ * E_;

  // Prefetch B tile 0 into buffer 0 (overlaps with A-tile load below).
  {
    float* Bbuf = Bs;
    #pragma unroll
    for (int i = 0; i < (CODES_PER_ITER * E_ / 4) / 128; ++i) {  // 16 issues
      int flat = i * 128 + tid;          // float4 index within tile
      int code = flat >> 6;              // 64 float4 per code row
      int e4   = flat & 63;
      ASYNC_COPY16(cbg + (size_t)code * E_ + e4 * 4, Bbuf + code * LDB + e4 * 4);
    }
  }

  if (prev_idx != nullptr && tid < ROWS) Pi[tid] = prev_idx[nbase + tid];
  for (int i = tid; i < NC_; i += 128) Cn[i] = cnorm[g * NC_ + i];
  __syncthreads();                       // Pi visible before A-tile load

  // A-tile load: coalesced along t, stored transposed As[t][e].
  #pragma unroll 4
  for (int it = 0; it < (ROWS * E_) / 128; ++it) {
    int flat = it * 128 + tid;
    int e  = flat >> 6;                  // 0..255
    int tt = flat & (ROWS - 1);          // 0..63
    float v = x[xbase + (size_t)e * T_ + tt];
    if (prev_cb != nullptr) {
      int pid = Pi[tt];
      v -= prev_cb[((size_t)g * NC_ + pid) * E_ + e];
    }
    As[tt * LDA + e] = v;
  }

  // WMMA fragment coordinates (16x16x4 f32):
  //   A: lane&15 = row, lane>>4 = K half; v2f = {A[row][k], A[row][k+1]}
  //   B: lane&15 = col(code), same K striping
  //   C/D: vgpr r -> row r + 8*(lane>>4), col = lane&15
  const int rowA = w * 16 + (lane & 15);
  const int kh   = lane >> 4;
  const int col  = lane & 15;

  float best[8];
  int   bidx[8];
  #pragma unroll
  for (int r = 0; r < 8; ++r) { best[r] = 3.4e38f; bidx[r] = 0; }

  for (int ct = 0; ct < NITER; ++ct) {
    __syncthreads();                     // all waves done with tile ct-1
    if (ct + 1 < NITER) {                // prefetch tile ct+1 into other buffer
      float* Bn = Bs + ((ct + 1) & 1) * CODES_PER_ITER * LDB;
      const float* gsrc = cbg + (size_t)(ct + 1) * CODES_PER_ITER * E_;
      #pragma unroll
      for (int i = 0; i < 16; ++i) {
        int flat = i * 128 + tid;
        int code = flat >> 6;
        int e4   = flat & 63;
        ASYNC_COPY16(gsrc + (size_t)code * E_ + e4 * 4, Bn + code * LDB + e4 * 4);
      }
      WAIT_ASYNC(16);                    // tile ct complete, ct+1 in flight
    } else {
      WAIT_ASYNC(0);
    }
    __syncthreads();                     // tile ct visible to all waves

    const float* Bbuf = Bs + (ct & 1) * CODES_PER_ITER * LDB;
    v8f acc0 = {};
    v8f acc1 = {};
    const float* ap  = As   + rowA * LDA + 2 * kh;
    const float* b0p = Bbuf + col * LDB + 2 * kh;
    const float* b1p = Bbuf + (col + 16) * LDB + 2 * kh;
    #pragma unroll
    for (int k0 = 0; k0 < E_; k0 += 4) {
      v2f a  = *(const v2f*)(ap  + k0);
      v2f b0 = *(const v2f*)(b0p + k0);
      v2f b1 = *(const v2f*)(b1p + k0);
      acc0 = __builtin_amdgcn_wmma_f32_16x16x4_f32(
          false, a, false, b0, (short)0, acc0, false, false);
      acc1 = __builtin_amdgcn_wmma_f32_16x16x4_f32(
          false, a, false, b1, (short)0, acc1, false, false);
    }

    int   code0 = ct * CODES_PER_ITER + col;
    float cn0   = Cn[code0];
    float cn1   = Cn[code0 + 16];
    #pragma unroll
    for (int r = 0; r < 8; ++r) {
      float d0 = cn0 - 2.0f * acc0[r];   // ||c||^2 - 2 x.c (||x||^2 row-const)
      float d1 = cn1 - 2.0f * acc1[r];
      if (d0 < best[r]) { best[r] = d0; bidx[r] = code0; }
      if (d1 < best[r]) { best[r] = d1; bidx[r] = code0 + 16; }
    }
  }

  // Argmin reduce across the 16 lanes of each half (xor<16 stays in-half);
  // tie-break on lower code index to match jnp.argmin first-occurrence.
  #pragma unroll
  for (int r = 0; r < 8; ++r) {
    float bd = best[r]; int bi = bidx[r];
    #pragma unroll
    for (int off = 1; off < 16; off <<= 1) {
      float od = __shfl_xor(bd, off, 32);
      int   oi = __shfl_xor(bi, off, 32);
      if (od < bd || (od == bd && oi < bi)) { bd = od; bi = oi; }
    }
    best[r] = bd; bidx[r] = bi;
  }

  if ((lane & 15) == 0) {                // lanes 0 (rows 0..7), 16 (rows 8..15)
    int mb = 8 * kh;
    #pragma unroll
    for (int r = 0; r < 8; ++r)
      idx_out[nbase + w * 16 + mb + r] = bidx[r];
  }
}

// ---------------------------------------------------------------------------
// Fused epilogue: gather z_q1/z_q2, write quantized_out, accumulate both MSEs.
__global__ void final_kernel(const float* __restrict__ x,
                             const float* __restrict__ cb1,
                             const float* __restrict__ cb2,
                             const int* __restrict__ idx1,
                             const int* __restrict__ idx2,
                             float* __restrict__ out,
                             float* __restrict__ sums) {
  size_t i = (size_t)blockIdx.x * blockDim.x + threadIdx.x;  // exact grid
  int t = (int)(i % T_);
  int c = (int)((i / T_) % C_);
  int b = (int)(i / ((size_t)C_ * T_));
  int g = c >> 8;
  int e = c & 255;
  size_t n = (size_t)g * N_ + (size_t)b * T_ + t;
  float z1 = cb1[((size_t)g * NC_ + idx1[n]) * E_ + e];
  float z2 = cb2[((size_t)g * NC_ + idx2[n]) * E_ + e];
  float xv = x[i];
  float r1 = xv - z1;                    // stage-2 input residual
  float d2 = z2 - r1;
  out[i] = z1 + z2;                      // straight-through forward value
  float s1 = r1 * r1;                    // (z_q1 - x)^2
  float s2 = d2 * d2;                    // (z_q2 - residual)^2
  #pragma unroll
  for (int off = 16; off > 0; off >>= 1) {
    s1 += __shfl_xor(s1, off, 32);
    s2 += __shfl_xor(s2, off, 32);
  }
  if ((threadIdx.x & 31) == 0) {
    atomicAdd(&sums[0], s1);
    atomicAdd(&sums[1], s2);
  }
}

// loss scalar + indices (as floats) into the output tail.
__global__ void finish_kernel(const float* __restrict__ sums,
                              const int* __restrict__ idx1,
                              const int* __restrict__ idx2,
                              float* __restrict__ out_tail) {
  size_t k = (size_t)blockIdx.x * blockDim.x + threadIdx.x;
  if (k == 0) {
    // loss = (1.25*mean1 + 1.25*mean2)/2 = 0.625*(s1+s2)/M
    out_tail[0] = 0.625f * (sums[0] + sums[1]) / (float)M_TOT;
  }
  if (k < (size_t)G_ * N_) {
    out_tail[1 + k] = (float)idx1[k];
    out_tail[1 + (size_t)G_ * N_ + k] = (float)idx2[k];
  }
}

// ---------------------------------------------------------------------------
extern "C" void kernel_launch(void* const* d_in, const int* in_sizes, int n_in,
                              void* d_out, int out_size, void* d_ws, size_t ws_size,
                              hipStream_t stream) {
  const float* xin = (const float*)d_in[0];
  const float* cb1 = (const float*)d_in[1];
  const float* cb2 = (const float*)d_in[2];
  float* out = (float*)d_out;

  // workspace layout (~541 KB)
  int*   idx1 = (int*)d_ws;                       // G*N ints
  int*   idx2 = idx1 + G_ * N_;                   // G*N ints
  float* cn1  = (float*)(idx2 + G_ * N_);         // G*NC floats
  float* cn2  = cn1 + G_ * NC_;                   // G*NC floats
  float* sums = cn2 + G_ * NC_;                   // 2 floats

  zero_kernel<<<1, 64, 0, stream>>>(sums);
  cnorm_kernel<<<(2 * G_ * NC_ * 32) / 256, 256, 0, stream>>>(cb1, cb2, cn1, cn2);

  size_t smem = ((size_t)ROWS * LDA + 2 * CODES_PER_ITER * LDB + NC_ + ROWS)
                * sizeof(float);                  // ~137.5 KB -> 2 blocks/WGP
  dim3 agrid(G_ * B_ * (T_ / ROWS));              // 1024 blocks, 128 threads
  argmin_kernel<<<agrid, 128, smem, stream>>>(xin, cb1, cn1, nullptr, nullptr, idx1);
  argmin_kernel<<<agrid, 128, smem, stream>>>(xin, cb2, cn2, cb1, idx1, idx2);

  final_kernel<<<(unsigned)(M_TOT / 256), 256, 0, stream>>>(
      xin, cb1, cb2, idx1, idx2, out, sums);
  finish_kernel<<<(G_ * N_) / 256, 256, 0, stream>>>(
      sums, idx1, idx2, out + M_TOT);
}